// DistributedKnowledgeCongruence_search_based_56410100465965
// MI455X (gfx1250) — compile-verified
//
#include <hip/hip_runtime.h>
#include <stdint.h>

// ---------------------------------------------------------------------------
// Per-row temperature search (60-step bisection on softmax entropy) + softmax.
// MI455X / gfx1250: wave32, one wave per row, row resident in VGPRs,
// TDM (tensor_load_to_lds) staging global->LDS, shfl_xor wave reductions,
// raw v_exp_f32 / v_log_f32 / v_rcp_f32, non-temporal streaming stores.
// ---------------------------------------------------------------------------

typedef unsigned int u32;
typedef u32   v4u __attribute__((ext_vector_type(4)));
typedef int   v8i __attribute__((ext_vector_type(8)));
typedef int   v4i __attribute__((ext_vector_type(4)));
typedef float f4  __attribute__((ext_vector_type(4)));

#define NCLS            1000
#define ROWS_PER_BLOCK  8
#define N_ITERS         60
#define TARGET_ENT      2.0f
#define EPS             1e-7f
#define LOG2E           1.44269504088896340736f

// Device-only fast ops (host pass just needs these to parse).
__device__ __forceinline__ float fast_exp2(float x) {
#if defined(__HIP_DEVICE_COMPILE__)
  return __builtin_amdgcn_exp2f(x);       // v_exp_f32 (base-2)
#else
  return x;
#endif
}
__device__ __forceinline__ float fast_log2(float x) {
#if defined(__HIP_DEVICE_COMPILE__)
  return __builtin_amdgcn_logf(x);        // v_log_f32 (base-2)
#else
  return x;
#endif
}
__device__ __forceinline__ float fast_rcp(float x) {
#if defined(__HIP_DEVICE_COMPILE__)
  return __builtin_amdgcn_rcpf(x);        // v_rcp_f32 (1 ulp)
#else
  return x;
#endif
}

__device__ __forceinline__ float wave_sum32(float v) {
#pragma unroll
  for (int m = 16; m >= 1; m >>= 1) v += __shfl_xor(v, m, 32);
  return v;
}

__device__ __forceinline__ float wave_max32(float v) {
#pragma unroll
  for (int m = 16; m >= 1; m >>= 1) v = fmaxf(v, __shfl_xor(v, m, 32));
  return v;
}

__global__ __launch_bounds__(256) void soften_rows_kernel(
    const float* __restrict__ logits, float* __restrict__ out, int nrows) {
  // 8 waves/block, each wave owns a 1024-float (4KB) LDS slot for its row.
  __shared__ __align__(16) float smem[ROWS_PER_BLOCK * 1024];

  const int lane = threadIdx.x & 31;
  const int wave = threadIdx.x >> 5;       // uniform within a wave
  const int row  = blockIdx.x * ROWS_PER_BLOCK + wave;
  if (row >= nrows) return;                // wave-uniform exit

  // Per-lane row slice: element index = c*128 + lane*4 + k, c in [0,8).
  // 1000 = 7*128 + 26*4  ->  chunk 7 valid only for lanes 0..25.
  float x[32];

#if defined(__HIP_DEVICE_COMPILE__) && \
    __has_builtin(__builtin_amdgcn_tensor_load_to_lds)
  {
    // --- Tensor Data Mover: DMA 1000 contiguous f32 -> this wave's LDS slot.
    // D# group0: count=1, lds_addr, global_addr[56:0], type=2 ("image").
    const uint64_t ga = (uint64_t)(uintptr_t)(logits + (size_t)row * NCLS);
    v4u g0 = { 1u,
               (u32)(wave * 4096),
               (u32)ga,
               (u32)(((ga >> 32) & 0x01FFFFFFu) | 0x80000000u) };
    // D# group1: wg_mask=0, data_size=2 (4B), tensor_dim0=1000, tensor_dim1=1,
    //            tile_dim0=1000, tile_dim1=1, tensor_dim0_stride=1000.
    v8i g1 = { (int)0x00020000,        // data_size=4B in bits[17:16]
               (int)(1000u << 16),     // tensor_dim0[15:0] << 16
               (int)(1u << 16),        // tensor_dim1[15:0] << 16
               (int)(1000u << 16),     // tile_dim0 << 16
               1,                      // tile_dim1 = 1
               1000,                   // tensor_dim0_stride[31:0]
               0, 0 };
    v4i gz4 = { 0, 0, 0, 0 };          // 2D tensor: groups 2/3 unused
    v8i gz8 = { 0, 0, 0, 0, 0, 0, 0, 0 };
    __builtin_amdgcn_tensor_load_to_lds(g0, g1, gz4, gz4, gz8, 0);
    __builtin_amdgcn_s_wait_tensorcnt(0);

    const float* rsm = &smem[wave * 1024];
    // The TDM intrinsic's LDS write isn't visible to LLVM's alias analysis;
    // escape the LDS pointer through an asm barrier so the compiler must
    // assume the DMA wrote through it (otherwise the ds_loads fold to undef).
    __asm__ volatile("" : : "r"(rsm) : "memory");
#pragma unroll
    for (int c = 0; c < 8; ++c) {      // ds_load_b128 x8 into registers
      f4 v = *(const f4*)(rsm + c * 128 + lane * 4);
      x[c * 4 + 0] = v[0]; x[c * 4 + 1] = v[1];
      x[c * 4 + 2] = v[2]; x[c * 4 + 3] = v[3];
    }
  }
#else
  {
    (void)smem;
    const float* rp = logits + (size_t)row * NCLS;
#pragma unroll
    for (int c = 0; c < 8; ++c) {
      if (c < 7 || lane < 26) {
        f4 v = *(const f4*)(rp + c * 128 + lane * 4);
        x[c * 4 + 0] = v[0]; x[c * 4 + 1] = v[1];
        x[c * 4 + 2] = v[2]; x[c * 4 + 3] = v[3];
      }
    }
  }
#endif
  if (lane >= 26) {                    // tail padding never contributes
    x[28] = x[29] = x[30] = x[31] = -__builtin_inff();
  }

  // Row max (numerical stability): d_i = x_i - max(x), exp2(d*k) <= 1.
  float mx = -__builtin_inff();
#pragma unroll
  for (int i = 0; i < 32; ++i) mx = fmaxf(mx, x[i]);
  mx = wave_max32(mx);
#pragma unroll
  for (int i = 0; i < 32; ++i) x[i] -= mx;

  // Entropy with eps, restructured (w = e_i + eps*Z, p+eps = w/Z):
  //   H = log2(Z)*(1 + NCLS*eps) - (sum w*log2 w)/Z
  // -> exactly 1 exp2 + 1 log2 per element per iteration.
  const float tmask = (lane < 26) ? 1.0f : 0.0f;
  float lo = 1e-4f, hi = 1000.0f;
  float e[32];

  for (int it = 0; it < N_ITERS; ++it) {
    const float mid = 0.5f * (lo + hi);
    const float k   = LOG2E * fast_rcp(mid);  // (x - m)/S in log2 units
    float sE = 0.0f;
#pragma unroll
    for (int i = 0; i < 32; ++i) {     // pass 1: e_i, partial Z
      e[i] = fast_exp2(x[i] * k);      // padded lanes: exp2(-inf)=0
      sE += e[i];
    }
    const float Z    = wave_sum32(sE);
    const float rcpZ = fast_rcp(Z);    // Z in [1, 1000]: rcp is safe
    const float ez   = EPS * Z;
    float sWL = 0.0f;
#pragma unroll
    for (int i = 0; i < 28; ++i) {     // pass 2: sum w*log2(w)
      const float w = e[i] + ez;
      sWL += w * fast_log2(w);
    }
#pragma unroll
    for (int i = 28; i < 32; ++i) {    // tail chunk masked per-lane
      const float w = e[i] + ez;
      sWL += tmask * (w * fast_log2(w));
    }
    const float WL = wave_sum32(sWL);
    const float H  = fast_log2(Z) * (1.0f + (float)NCLS * EPS) - WL * rcpZ;
    const bool cold = (H < TARGET_ENT);  // entropy below target -> warmer S
    lo = cold ? mid : lo;                // wave-uniform: no divergence
    hi = cold ? hi : mid;
  }

  // Final congruent distribution: softmax(x / S), no eps.
  const float S = 0.5f * (lo + hi);
  const float kf = LOG2E / S;          // precise for the output pass
  float sE = 0.0f;
#pragma unroll
  for (int i = 0; i < 32; ++i) {
    e[i] = fast_exp2(x[i] * kf);
    sE += e[i];
  }
  const float invZ = 1.0f / wave_sum32(sE);  // precise normalization

  float* op = out + (size_t)row * NCLS;
#pragma unroll
  for (int c = 0; c < 8; ++c) {
    if (c < 7 || lane < 26) {
      f4 q;
      q[0] = e[c * 4 + 0] * invZ; q[1] = e[c * 4 + 1] * invZ;
      q[2] = e[c * 4 + 2] * invZ; q[3] = e[c * 4 + 3] * invZ;
      __builtin_nontemporal_store(q, (f4*)(op + c * 128 + lane * 4));
    }
  }
}

extern "C" void kernel_launch(void* const* d_in, const int* in_sizes, int n_in,
                              void* d_out, int out_size, void* d_ws, size_t ws_size,
                              hipStream_t stream) {
  const float* logits = (const float*)d_in[0];
  float* out = (float*)d_out;
  const int nrows = in_sizes[0] / NCLS;                      // 32768
  const int grid  = (nrows + ROWS_PER_BLOCK - 1) / ROWS_PER_BLOCK;
  soften_rows_kernel<<<dim3(grid), dim3(256), 0, stream>>>(logits, out, nrows);
}